// Predictor_67585605370461
// MI455X (gfx1250) — compile-verified
//
#include <hip/hip_runtime.h>

// ---------------------------------------------------------------------------
// MI455X (gfx1250, wave32) implementation.
// Memory-bound (~330 MB traffic => ~14us at 23.3 TB/s). WMMA f32 16x16x4 is
// used as the wave32 cross-lane reduction engine for the global avgB/avgW
// accumulators (a correct use of the documented A/D VGPR layouts).
// ---------------------------------------------------------------------------

typedef __attribute__((ext_vector_type(2))) float v2f;
typedef __attribute__((ext_vector_type(8))) float v8f;

#define BLOCK 256
#define IPT   8
#define NWAVE (BLOCK / 32)

// Full 32-lane sum via V_WMMA_F32_16X16X4_F32.
// A layout (16x4 f32): VGPR0 = {lanes 0-15: K=0, lanes 16-31: K=2},
//                      VGPR1 = {lanes 0-15: K=1, lanes 16-31: K=3}.
// With a.y = 0 and B = all-ones: D[m][n] = x(lane m) + x(lane m+16).
// D layout: VGPR v holds row M=v in lanes 0-15 and row M=v+8 in lanes 16-31,
// so sum(d.s0..d.s7) gives the half-wave sums; one xor-16 shuffle completes.
__device__ __forceinline__ float wave_sum32_wmma(float x) {
  v2f a; a.x = x;    a.y = 0.0f;
  v2f b; b.x = 1.0f; b.y = 1.0f;       // B = ones => its layout is immaterial
  v8f c = {};
  v8f d = __builtin_amdgcn_wmma_f32_16x16x4_f32(
      /*neg_a=*/false, a, /*neg_b=*/false, b,
      /*c_mod=*/(short)0, c, /*reuse_a=*/false, /*reuse_b=*/false);
  float t = ((d.s0 + d.s1) + (d.s2 + d.s3)) + ((d.s4 + d.s5) + (d.s6 + d.s7));
  t += __shfl_xor(t, 16, 32);          // combine the two half-wave sums
  return t;                            // full wave sum, broadcast to all lanes
}

__global__ void pred_init_acc(float* __restrict__ acc) {
  if (threadIdx.x < 4) acc[threadIdx.x] = 0.0f;
}

// Pass 1: per-patch classification + global {sumB,cntB,sumW,cntW} reduction.
__global__ __launch_bounds__(BLOCK) void pred_pass1(
    const float* __restrict__ image, const float* __restrict__ edges,
    float* __restrict__ out, float* __restrict__ acc, int Np, int Nout) {
  const int base = blockIdx.x * (BLOCK * IPT) + threadIdx.x;

  float sB = 0.0f, cB = 0.0f, sW = 0.0f, cW = 0.0f;

  for (int k = 0; k < IPT; ++k) {
    const int n = base + k * BLOCK;
    if (k + 1 < IPT) {  // speculative prefetch of next tile (global_prefetch_b8)
      __builtin_prefetch(image + (size_t)(base + (k + 1) * BLOCK) * 9, 0, 0);
      __builtin_prefetch(edges + (size_t)(base + (k + 1) * BLOCK) * 9, 0, 0);
    }
    if (n < Np) {
      const float* __restrict__ im = image + (size_t)n * 9;
      const float* __restrict__ ed = edges + (size_t)n * 9;
      int   cd = 0, ca = 0;
      float sd = 0.0f, sa = 0.0f;
      float ctr = 0.0f;
#pragma unroll
      for (int i = 0; i < 9; ++i) {
        const float e = ed[i];
        const float v = im[i];
        const bool dis = (e > 0.5f);
        const bool acm = (e != 0.0f) && !dis;
        cd += dis ? 1 : 0;
        ca += acm ? 1 : 0;
        sd += dis ? v : 0.0f;
        sa += acm ? v : 0.0f;
        if (i == 4) ctr = v;           // image[:,1,1]
      }
      const float md = sd / fmaxf((float)cd, 1.0f);
      const float ma = sa / fmaxf((float)ca, 1.0f);
      const float o  = ((cd > 0) && (ca > 0)) ? ((md > ma) ? 0.0f : 1.0f)
                                              : 2.0f;
      if (n < Nout) out[n] = o;
      if (o == 0.0f)      { sB += ctr; cB += 1.0f; }
      else if (o == 1.0f) { sW += ctr; cW += 1.0f; }
    }
  }

  // Wave-level reduction on the matrix pipe (EXEC is all-ones here: no
  // divergence at this point, grid covers whole domain with full blocks).
  sB = wave_sum32_wmma(sB);
  cB = wave_sum32_wmma(cB);
  sW = wave_sum32_wmma(sW);
  cW = wave_sum32_wmma(cW);

  __shared__ float lds[NWAVE][4];
  const int lane = threadIdx.x & 31;
  const int wv   = threadIdx.x >> 5;
  if (lane == 0) {
    lds[wv][0] = sB; lds[wv][1] = cB; lds[wv][2] = sW; lds[wv][3] = cW;
  }
  __syncthreads();
  if (threadIdx.x < 4) {
    float t = 0.0f;
#pragma unroll
    for (int w = 0; w < NWAVE; ++w) t += lds[w][threadIdx.x];
    atomicAdd(&acc[threadIdx.x], t);   // global_atomic_add_f32, 4 per block
  }
}

// Pass 2: resolve the out==2 patches against the global averages.
__global__ void pred_pass2(const float* __restrict__ image,
                           const float* __restrict__ acc,
                           float* __restrict__ out, int N) {
  const int n = blockIdx.x * blockDim.x + threadIdx.x;
  if (n >= N) return;
  const float o = out[n];
  if (o == 2.0f) {
    const float v    = image[(size_t)n * 9 + 4];
    const float avgB = acc[0] / fmaxf(acc[1], 1.0f);
    const float avgW = acc[2] / fmaxf(acc[3], 1.0f);
    out[n] = (fabsf(v - avgB) < fabsf(v - avgW)) ? 0.0f : 1.0f;
  }
}

extern "C" void kernel_launch(void* const* d_in, const int* in_sizes, int n_in,
                              void* d_out, int out_size, void* d_ws, size_t ws_size,
                              hipStream_t stream) {
  (void)n_in; (void)ws_size;
  const float* image = (const float*)d_in[0];   // [N,3,3] f32
  const float* edges = (const float*)d_in[1];   // [N,3,3] f32
  // d_in[2] (gt) is used by the reference only for the output shape.
  float* out = (float*)d_out;                   // [2048*2048] f32
  float* acc = (float*)d_ws;                    // {sumB,cntB,sumW,cntW}

  const int Np   = in_sizes[0] / 9;             // number of patches
  const int Nout = out_size;                    // == Np for this problem

  pred_init_acc<<<1, 32, 0, stream>>>(acc);

  const int tiles = (Np + BLOCK * IPT - 1) / (BLOCK * IPT);
  pred_pass1<<<tiles, BLOCK, 0, stream>>>(image, edges, out, acc, Np, Nout);

  pred_pass2<<<(Nout + 255) / 256, 256, 0, stream>>>(image, acc, out, Nout);
}